// AffinityNet_75557064671580
// MI455X (gfx1250) — compile-verified
//
#include <hip/hip_runtime.h>
#include <hip/hip_bf16.h>
#include <math.h>

// ---------------------------------------------------------------------------
// GATv2 x2 + mean-pool + MLP head for MI455X (gfx1250, wave32).
//  - Node GEMMs use V_WMMA_F32_16X16X4_F32 (native fp32 WMMA, keeps reference
//    precision, one wave per 16x16 output tile).
//  - Edge sweeps: 16 lanes/edge, float4 (b128) gathers that hit the 192MB L2
//    (node tables are 12.8MB), shuffle-reduced attention dot, fp32 atomics for
//    segment sum, ordered-uint atomicMax for segment max.
// ---------------------------------------------------------------------------

typedef __attribute__((ext_vector_type(2))) float v2f;
typedef __attribute__((ext_vector_type(8))) float v8f;

#define N_NODES   50000
#define N_EDGES   1600000
#define EP        (N_EDGES + N_NODES)   // edges + self loops
#define G_GRAPHS  64
#define C_IN      128
#define C_HID     64
#define NEG_SLOPE 0.2f

// ---- ordered-float <-> uint encoding for atomic segment max ---------------
__device__ __forceinline__ unsigned enc_f32(float f) {
    unsigned u = __float_as_uint(f);
    return (u & 0x80000000u) ? ~u : (u | 0x80000000u);
}
__device__ __forceinline__ float dec_f32(unsigned u) {
    unsigned b = (u & 0x80000000u) ? (u & 0x7FFFFFFFu) : ~u;
    return __uint_as_float(b);
}

// ---- self-loop attr = mean of incoming edge attrs --------------------------
__global__ void k_loop_accum(const int* __restrict__ dst, const float* __restrict__ ea,
                             float* __restrict__ sum, float* __restrict__ cnt) {
    int e = blockIdx.x * blockDim.x + threadIdx.x;
    if (e >= N_EDGES) return;
    int d = dst[e];
    atomicAdd(sum + d, ea[e]);
    atomicAdd(cnt + d, 1.0f);
}
__global__ void k_loop_final(const float* __restrict__ sum, const float* __restrict__ cnt,
                             float* __restrict__ la) {
    int n = blockIdx.x * blockDim.x + threadIdx.x;
    if (n >= N_NODES) return;
    la[n] = sum[n] / fmaxf(cnt[n], 1.0f);
}

// ---- fp32 WMMA GEMM: Y(nrows x 64) = X(nrows x kin) @ W(kin x 64) + bias ---
// block = 128 threads (4 waves); wave w owns columns [16w,16w+16); blockIdx.x
// owns rows [16b,16b+16). K stepped by 4 with V_WMMA_F32_16X16X4_F32.
__global__ __launch_bounds__(128)
void k_gemm_bias_wmma(const float* __restrict__ X, const float* __restrict__ W,
                      const float* __restrict__ bias, float* __restrict__ Y,
                      int nrows, int kin) {
    const int lane = threadIdx.x & 31;
    const int wave = threadIdx.x >> 5;       // 0..3 -> column tile
    const int row0 = blockIdx.x * 16;
    const int col0 = wave * 16;
    const int m    = lane & 15;              // A row / B col / C col within tile
    const int kh   = (lane >> 4) * 2;        // lane half selects K pair {0,1} / {2,3}

    int arow = row0 + m;
    if (arow >= nrows) arow = nrows - 1;     // N is a multiple of 16; clamp is a no-op
    const float* xrow = X + (size_t)arow * kin;

    v8f acc = {0.f, 0.f, 0.f, 0.f, 0.f, 0.f, 0.f, 0.f};
    for (int k = 0; k < kin; k += 4) {
        v2f a, b;
        a[0] = xrow[k + kh];
        a[1] = xrow[k + kh + 1];
        b[0] = W[(k + kh)     * C_HID + col0 + m];
        b[1] = W[(k + kh + 1) * C_HID + col0 + m];
        acc = __builtin_amdgcn_wmma_f32_16x16x4_f32(
            /*neg_a=*/false, a, /*neg_b=*/false, b,
            /*c_mod=*/(short)0, acc, /*reuse_a=*/false, /*reuse_b=*/false);
    }

    const int col   = col0 + m;
    const float bv  = bias[col];
    const int rbase = row0 + (lane >> 4) * 8;   // C layout: VGPR r -> row r / r+8
#pragma unroll
    for (int r = 0; r < 8; ++r) {
        int rr = rbase + r;
        if (rr < nrows) Y[(size_t)rr * C_HID + col] = acc[r] + bv;
    }
}

// ---- init accumulator to bias ---------------------------------------------
__global__ void k_fill_bias(float* __restrict__ out, const float* __restrict__ bias) {
    int i = blockIdx.x * blockDim.x + threadIdx.x;
    if (i >= N_NODES * C_HID) return;
    out[i] = bias[i & (C_HID - 1)];
}

// ---- edge pass 1: logits + segment max (16 lanes/edge, 4 channels each) ----
__global__ void k_edge_logits(const float* __restrict__ xl, const float* __restrict__ xr,
                              const int* __restrict__ src, const int* __restrict__ dst,
                              const float* __restrict__ ea, const float* __restrict__ la,
                              const float* __restrict__ We, const float* __restrict__ att,
                              float* __restrict__ logits, unsigned* __restrict__ menc) {
    long long t = (long long)blockIdx.x * blockDim.x + threadIdx.x;
    long long e = t >> 4;
    if (e >= EP) return;
    const int g  = (int)(t & 15);
    const int c0 = g * 4;
    int s, d; float a;
    if (e < N_EDGES) { s = src[e]; d = dst[e]; a = ea[e]; }
    else             { s = d = (int)(e - N_EDGES); a = la[s]; }

    float4 vl = *(const float4*)(xl + (size_t)s * C_HID + c0);
    float4 vr = *(const float4*)(xr + (size_t)d * C_HID + c0);
    float4 vw = *(const float4*)(We + c0);
    float4 va = *(const float4*)(att + c0);

    float p = 0.f, x;
    x = vl.x + vr.x + a * vw.x; x = x > 0.f ? x : NEG_SLOPE * x; p += x * va.x;
    x = vl.y + vr.y + a * vw.y; x = x > 0.f ? x : NEG_SLOPE * x; p += x * va.y;
    x = vl.z + vr.z + a * vw.z; x = x > 0.f ? x : NEG_SLOPE * x; p += x * va.z;
    x = vl.w + vr.w + a * vw.w; x = x > 0.f ? x : NEG_SLOPE * x; p += x * va.w;

    p += __shfl_xor(p, 8, 16);
    p += __shfl_xor(p, 4, 16);
    p += __shfl_xor(p, 2, 16);
    p += __shfl_xor(p, 1, 16);

    if (g == 0) {
        logits[e] = p;
        atomicMax(menc + d, enc_f32(p));
    }
}

__global__ void k_decode_max(const unsigned* __restrict__ menc, float* __restrict__ m) {
    int n = blockIdx.x * blockDim.x + threadIdx.x;
    if (n >= N_NODES) return;
    unsigned u = menc[n];
    float v = dec_f32(u);
    m[n] = (u == 0u || !isfinite(v)) ? 0.f : v;   // reference: non-finite max -> 0
}

// ---- edge pass 2: p = exp(logit - m[dst]); denom[dst] += p (in place) ------
__global__ void k_edge_exp(float* __restrict__ p, const float* __restrict__ m,
                           const int* __restrict__ dst, float* __restrict__ denom) {
    long long e = (long long)blockIdx.x * blockDim.x + threadIdx.x;
    if (e >= EP) return;
    int d = (e < N_EDGES) ? dst[e] : (int)(e - N_EDGES);
    float v = expf(p[e] - m[d]);
    p[e] = v;
    atomicAdd(denom + d, v);
}

__global__ void k_invdenom(float* __restrict__ denom) {
    int n = blockIdx.x * blockDim.x + threadIdx.x;
    if (n >= N_NODES) return;
    denom[n] = 1.0f / (denom[n] + 1e-16f);
}

// ---- edge pass 3: out[dst] += alpha * xl[src] ------------------------------
__global__ void k_edge_accum(const float* __restrict__ p, const float* __restrict__ invd,
                             const float* __restrict__ xl,
                             const int* __restrict__ src, const int* __restrict__ dst,
                             float* __restrict__ out) {
    long long t = (long long)blockIdx.x * blockDim.x + threadIdx.x;
    long long e = t >> 4;
    if (e >= EP) return;
    const int c0 = (int)(t & 15) * 4;
    int s, d;
    if (e < N_EDGES) { s = src[e]; d = dst[e]; }
    else             { s = d = (int)(e - N_EDGES); }
    float alpha = p[e] * invd[d];
    float4 vl = *(const float4*)(xl + (size_t)s * C_HID + c0);
    float* o = out + (size_t)d * C_HID + c0;
    atomicAdd(o + 0, alpha * vl.x);
    atomicAdd(o + 1, alpha * vl.y);
    atomicAdd(o + 2, alpha * vl.z);
    atomicAdd(o + 3, alpha * vl.w);
}

__global__ void k_elu(float* __restrict__ h) {
    int i = blockIdx.x * blockDim.x + threadIdx.x;
    if (i >= N_NODES * C_HID) return;
    float x = h[i];
    h[i] = x > 0.f ? x : expm1f(x);
}

// ---- global mean pool ------------------------------------------------------
__global__ void k_pool(const float* __restrict__ h, const int* __restrict__ batch,
                       float* __restrict__ pooled, float* __restrict__ gcnt) {
    long long t = (long long)blockIdx.x * blockDim.x + threadIdx.x;
    long long n = t >> 4;
    if (n >= N_NODES) return;
    const int g16 = (int)(t & 15);
    const int c0  = g16 * 4;
    int g = batch[n];
    float4 v = *(const float4*)(h + (size_t)n * C_HID + c0);
    float* o = pooled + (size_t)g * C_HID + c0;
    atomicAdd(o + 0, v.x);
    atomicAdd(o + 1, v.y);
    atomicAdd(o + 2, v.z);
    atomicAdd(o + 3, v.w);
    if (g16 == 0) atomicAdd(gcnt + g, 1.0f);
}

// ---- head: relu(fc1) -> BN(eval) -> fc3 ; one block per graph --------------
__global__ __launch_bounds__(32)
void k_head(const float* __restrict__ pooled, const float* __restrict__ gcnt,
            const float* __restrict__ W1, const float* __restrict__ b1,
            const float* __restrict__ gamma, const float* __restrict__ beta,
            const float* __restrict__ mean, const float* __restrict__ var,
            const float* __restrict__ W3, const float* __restrict__ b3,
            float* __restrict__ out) {
    const int g = blockIdx.x;
    const int j = threadIdx.x;   // 0..31 hidden units
    float inv = 1.0f / fmaxf(gcnt[g], 1.0f);
    float z = b1[j];
#pragma unroll 8
    for (int c = 0; c < C_HID; ++c)
        z += pooled[g * C_HID + c] * inv * W1[c * 32 + j];
    z = fmaxf(z, 0.f);
    z = (z - mean[j]) * rsqrtf(var[j] + 1e-5f) * gamma[j] + beta[j];
    float t = z * W3[j];
    t += __shfl_xor(t, 16, 32);
    t += __shfl_xor(t, 8, 32);
    t += __shfl_xor(t, 4, 32);
    t += __shfl_xor(t, 2, 32);
    t += __shfl_xor(t, 1, 32);
    if (j == 0) out[g] = t + b3[0];
}

// ---------------------------------------------------------------------------
extern "C" void kernel_launch(void* const* d_in, const int* in_sizes, int n_in,
                              void* d_out, int out_size, void* d_ws, size_t ws_size,
                              hipStream_t stream) {
    const float* x     = (const float*)d_in[0];
    const int*   eidx  = (const int*)d_in[1];
    const float* eattr = (const float*)d_in[2];
    const int*   batch = (const int*)d_in[3];
    const float *Wl1=(const float*)d_in[4],  *bl1=(const float*)d_in[5];
    const float *Wr1=(const float*)d_in[6],  *br1=(const float*)d_in[7];
    const float *We1=(const float*)d_in[8],  *att1=(const float*)d_in[9], *bias1=(const float*)d_in[10];
    const float *Wl2=(const float*)d_in[11], *bl2=(const float*)d_in[12];
    const float *Wr2=(const float*)d_in[13], *br2=(const float*)d_in[14];
    const float *We2=(const float*)d_in[15], *att2=(const float*)d_in[16], *bias2=(const float*)d_in[17];
    const float *Wfc1=(const float*)d_in[18], *bfc1=(const float*)d_in[19];
    const float *bng=(const float*)d_in[20], *bnb=(const float*)d_in[21];
    const float *bnm=(const float*)d_in[22], *bnv=(const float*)d_in[23];
    const float *Wfc3=(const float*)d_in[24], *bfc3=(const float*)d_in[25];

    const int* src = eidx;
    const int* dst = eidx + N_EDGES;

    // ---- workspace layout (floats) ----
    const size_t N64 = (size_t)N_NODES * C_HID;
    float*    A    = (float*)d_ws;           // xl   (N x 64)
    float*    B    = A + N64;                // xr   (N x 64)
    float*    C    = B + N64;                // accum / h (N x 64)
    float*    P    = C + N64;                // per-edge logits -> p (EP)
    float*    lsum = P + EP;                 // N
    float*    lcnt = lsum + N_NODES;         // N (contiguous with lsum)
    float*    lattr= lcnt + N_NODES;         // N
    unsigned* menc = (unsigned*)(lattr + N_NODES);     // N
    float*    mval = (float*)(menc + N_NODES);         // N
    float*    denom= mval + N_NODES;                   // N
    float*    pooled = denom + N_NODES;                // G x 64
    float*    gcnt   = pooled + (size_t)G_GRAPHS * C_HID; // G (contiguous)

    const int TB = 256;
    const int gN    = (N_NODES + TB - 1) / TB;
    const int gE    = (N_EDGES + TB - 1) / TB;
    const int gEP   = (EP + TB - 1) / TB;
    const int gEP16 = (int)(((long long)EP * 16 + TB - 1) / TB);
    const int gN64  = (int)((N64 + TB - 1) / TB);
    const int gN16  = (int)(((long long)N_NODES * 16 + TB - 1) / TB);
    const int gGemm = (N_NODES + 15) / 16;   // 3125 row tiles

    // ---- self-loop attrs (once, shared by both layers) ----
    hipMemsetAsync(lsum, 0, 2 * (size_t)N_NODES * sizeof(float), stream);
    k_loop_accum<<<gE, TB, 0, stream>>>(dst, eattr, lsum, lcnt);
    k_loop_final<<<gN, TB, 0, stream>>>(lsum, lcnt, lattr);

    // ================= GATv2 layer 1 (in=128) =================
    k_gemm_bias_wmma<<<gGemm, 128, 0, stream>>>(x, Wl1, bl1, A, N_NODES, C_IN);
    k_gemm_bias_wmma<<<gGemm, 128, 0, stream>>>(x, Wr1, br1, B, N_NODES, C_IN);
    hipMemsetAsync(menc,  0, (size_t)N_NODES * sizeof(unsigned), stream);
    hipMemsetAsync(denom, 0, (size_t)N_NODES * sizeof(float), stream);
    k_fill_bias<<<gN64, TB, 0, stream>>>(C, bias1);
    k_edge_logits<<<gEP16, TB, 0, stream>>>(A, B, src, dst, eattr, lattr, We1, att1, P, menc);
    k_decode_max<<<gN, TB, 0, stream>>>(menc, mval);
    k_edge_exp<<<gEP, TB, 0, stream>>>(P, mval, dst, denom);
    k_invdenom<<<gN, TB, 0, stream>>>(denom);
    k_edge_accum<<<gEP16, TB, 0, stream>>>(P, denom, A, src, dst, C);
    k_elu<<<gN64, TB, 0, stream>>>(C);                 // C = h1

    // ================= GATv2 layer 2 (in=64) =================
    k_gemm_bias_wmma<<<gGemm, 128, 0, stream>>>(C, Wl2, bl2, A, N_NODES, C_HID);
    k_gemm_bias_wmma<<<gGemm, 128, 0, stream>>>(C, Wr2, br2, B, N_NODES, C_HID);
    hipMemsetAsync(menc,  0, (size_t)N_NODES * sizeof(unsigned), stream);
    hipMemsetAsync(denom, 0, (size_t)N_NODES * sizeof(float), stream);
    k_fill_bias<<<gN64, TB, 0, stream>>>(C, bias2);    // h1 already consumed by GEMMs
    k_edge_logits<<<gEP16, TB, 0, stream>>>(A, B, src, dst, eattr, lattr, We2, att2, P, menc);
    k_decode_max<<<gN, TB, 0, stream>>>(menc, mval);
    k_edge_exp<<<gEP, TB, 0, stream>>>(P, mval, dst, denom);
    k_invdenom<<<gN, TB, 0, stream>>>(denom);
    k_edge_accum<<<gEP16, TB, 0, stream>>>(P, denom, A, src, dst, C);
    k_elu<<<gN64, TB, 0, stream>>>(C);                 // C = h2

    // ================= pool + head =================
    hipMemsetAsync(pooled, 0, ((size_t)G_GRAPHS * C_HID + G_GRAPHS) * sizeof(float), stream);
    k_pool<<<gN16, TB, 0, stream>>>(C, batch, pooled, gcnt);
    k_head<<<G_GRAPHS, 32, 0, stream>>>(pooled, gcnt, Wfc1, bfc1, bng, bnb, bnm, bnv,
                                        Wfc3, bfc3, (float*)d_out);
}